// HeteroHGNN_32401233281387
// MI455X (gfx1250) — compile-verified
//
#include <hip/hip_runtime.h>
#include <math.h>

#define D 128
#define TPB 256
#define WAVES_PER_BLOCK 8   // 256 threads / wave32

typedef __attribute__((ext_vector_type(2))) float v2f;
typedef __attribute__((ext_vector_type(8))) float v8f;

__device__ __forceinline__ void atom_add_f32(float* p, float v) {
  // lowers to GLOBAL_ATOMIC_ADD_F32 (no return) at device scope
  __hip_atomic_fetch_add(p, v, __ATOMIC_RELAXED, __HIP_MEMORY_SCOPE_AGENT);
}

// ---------------- degree / normalization ----------------

__global__ void k_init_deg(float* dp, float* dc, float* dr, int n) {
  int i = blockIdx.x * blockDim.x + threadIdx.x;
  if (i < n) { dp[i] = 1.0f; dc[i] = 1.0f; dr[i] = 1.0f; }  // self-loop contributes 1
}

__global__ void k_count_deg(const int* __restrict__ ei, int E, float* __restrict__ deg) {
  int e = blockIdx.x * blockDim.x + threadIdx.x;
  if (e < E) atom_add_f32(&deg[ei[E + e]], 1.0f);   // dst = ei[1][e]
}

__global__ void k_dinv(float* dp, float* dc, float* dr, int n) {
  int i = blockIdx.x * blockDim.x + threadIdx.x;
  if (i < n) {
    dp[i] = rsqrtf(dp[i]);   // deg >= 1 always (self-loop)
    dc[i] = rsqrtf(dc[i]);
    dr[i] = rsqrtf(dr[i]);
  }
}

// per-edge symmetric norm, computed once and reused by both layers
__global__ void k_norm(const int* __restrict__ ei, int E,
                       const float* __restrict__ dinv, float* __restrict__ norm) {
  int e = blockIdx.x * blockDim.x + threadIdx.x;
  if (e < E) norm[e] = dinv[ei[e]] * dinv[ei[E + e]];
}

__global__ void k_zero4(float4* __restrict__ a, int n4) {
  int i = blockIdx.x * blockDim.x + threadIdx.x;
  if (i < n4) a[i] = make_float4(0.f, 0.f, 0.f, 0.f);
}

// ---------------- fp32 WMMA GEMM + fused self-loop ----------------
// H[n x 128] = X[n x 128] @ W[128 x 128];  acc[row] += H[row] * dinv[row]^2
// One wave computes a 16-row tile; W staged in LDS once per block.
// Single accumulation chain: WMMA D->C chaining is the pipelined pattern on CDNA5
// (hazard table only penalizes D->A/B). n-loop kept rolled to avoid VGPR spills.
// V_WMMA_F32_16X16X4_F32 register layouts (ISA 7.12.2):
//   A 16x4 : lanes 0-15 -> M=lane, v0=K0,v1=K1 ; lanes 16-31 -> M=lane-16, v0=K2,v1=K3
//   B 4x16 : vj, lanes 0-15 -> row K=j ; lanes 16-31 -> row K=j+2 ; col N=lane&15
//   C/D    : vj, lanes 0-15 -> row M=j ; lanes 16-31 -> row M=j+8 ; col N=lane&15
__global__ void k_gemm_wmma(const float* __restrict__ X, const float* __restrict__ W,
                            const float* __restrict__ dinv,
                            float* __restrict__ H, float* __restrict__ acc, int nrows) {
  __shared__ float sW[D * D];                    // 64 KB of the WGP's 320 KB LDS
  {                                              // vectorized cooperative stage of W
    const float4* src4 = (const float4*)W;
    float4* dst4 = (float4*)sW;
    for (int i = threadIdx.x; i < D * D / 4; i += blockDim.x) dst4[i] = src4[i];
  }
  __syncthreads();

  const int wave = threadIdx.x >> 5;             // wave32
  const int lane = threadIdx.x & 31;
  const int half = lane >> 4;                    // 0: K pair {0,1}, 1: K pair {2,3}
  const int l16  = lane & 15;
  const int tile = blockIdx.x * WAVES_PER_BLOCK + wave;
  const int row0 = tile * 16;
  if (row0 >= nrows) return;                     // wave-uniform exit, EXEC stays all-1

  // Preload all A fragments for this 16-row strip: a[k] covers K = 4k..4k+3
  v2f a[32];
  const float* xrow = X + (size_t)(row0 + l16) * D + 2 * half;
#pragma unroll
  for (int k = 0; k < 32; ++k)
    a[k] = *(const v2f*)(xrow + 4 * k);

  // self-loop scale for the 8 rows this lane-half writes (uniform across l16)
  float sl[8];
#pragma unroll
  for (int j = 0; j < 8; ++j) {
    float dv = dinv[row0 + j + 8 * half];
    sl[j] = dv * dv;
  }

#pragma unroll 1                                 // keep rolled: avoids VGPR spills
  for (int n = 0; n < 8; ++n) {                  // 8 column tiles of 16
    const int n0 = n * 16;
    v8f c = {};
#pragma unroll
    for (int k = 0; k < 32; ++k) {
      // pair (rows 4k+2h, 4k+1+2h) fuses into one ds_load_2addr_stride64_b32
      v2f b;
      b.x = sW[(4 * k + 0 + 2 * half) * D + n0 + l16];
      b.y = sW[(4 * k + 1 + 2 * half) * D + n0 + l16];
      c = __builtin_amdgcn_wmma_f32_16x16x4_f32(
          /*neg_a=*/false, a[k], /*neg_b=*/false, b,
          /*c_mod=*/(short)0, c, /*reuse_a=*/false, /*reuse_b=*/false);
    }

    float* hp = H   + (size_t)(row0 + 8 * half) * D + n0 + l16;
    float* ap = acc + (size_t)(row0 + 8 * half) * D + n0 + l16;
#pragma unroll
    for (int j = 0; j < 8; ++j) {
      hp[(size_t)j * D] = c[j];
      ap[(size_t)j * D] += c[j] * sl[j];         // fused self-loop (row uniquely owned)
    }
  }
}

// ---------------- edge scatter: acc[dst] += h[src] * norm[e] ----------------
// 32 lanes per edge, float4 per lane (32*4 = 128 features). h/acc are L2-resident (51 MB each).
__global__ void k_scatter(const int* __restrict__ ei, int E,
                          const float* __restrict__ h, const float* __restrict__ norm,
                          float* __restrict__ acc) {
  long long t = (long long)blockIdx.x * blockDim.x + threadIdx.x;
  int e = (int)(t >> 5);
  int lane = (int)(t & 31);
  if (e >= E) return;
  int src = ei[e];
  int dst = ei[E + e];
  float nm = norm[e];
  const float4 v = *(const float4*)(h + (size_t)src * D + lane * 4);
  float* ap = acc + (size_t)dst * D + lane * 4;
  atom_add_f32(ap + 0, v.x * nm);
  atom_add_f32(ap + 1, v.y * nm);
  atom_add_f32(ap + 2, v.z * nm);
  atom_add_f32(ap + 3, v.w * nm);
}

// ---------------- layer finalization ----------------

__global__ void k_finalize1(const float* __restrict__ x, const float* __restrict__ acc,
                            const float* __restrict__ b1, const float* __restrict__ b2,
                            const float* __restrict__ b3, float* __restrict__ h1, int n4) {
  int t = blockIdx.x * blockDim.x + threadIdx.x;
  if (t >= n4) return;
  const int d0 = (t & 31) * 4;                   // 32 float4-groups per 128-wide row
  const float4 a  = ((const float4*)acc)[t];
  const float4 xi = ((const float4*)x)[t];
  const float inv3 = 1.0f / 3.0f;
  float bs[4];
#pragma unroll
  for (int j = 0; j < 4; ++j) bs[j] = b1[d0 + j] + b2[d0 + j] + b3[d0 + j];
  float v[4] = { xi.x + (a.x + bs[0]) * inv3, xi.y + (a.y + bs[1]) * inv3,
                 xi.z + (a.z + bs[2]) * inv3, xi.w + (a.w + bs[3]) * inv3 };
#pragma unroll
  for (int j = 0; j < 4; ++j) v[j] = v[j] >= 0.f ? v[j] : 0.01f * v[j];  // leaky_relu
  ((float4*)h1)[t] = make_float4(v[0], v[1], v[2], v[3]);
}

__global__ void k_finalize2(const float* __restrict__ acc,
                            const float* __restrict__ b1, const float* __restrict__ b2,
                            const float* __restrict__ b3, float* __restrict__ out, int n4) {
  int t = blockIdx.x * blockDim.x + threadIdx.x;
  if (t >= n4) return;
  const int d0 = (t & 31) * 4;
  const float4 a = ((const float4*)acc)[t];
  const float inv3 = 1.0f / 3.0f;
  float4 o;
  o.x = (a.x + b1[d0 + 0] + b2[d0 + 0] + b3[d0 + 0]) * inv3;
  o.y = (a.y + b1[d0 + 1] + b2[d0 + 1] + b3[d0 + 1]) * inv3;
  o.z = (a.z + b1[d0 + 2] + b2[d0 + 2] + b3[d0 + 2]) * inv3;
  o.w = (a.w + b1[d0 + 3] + b2[d0 + 3] + b3[d0 + 3]) * inv3;
  ((float4*)out)[t] = o;
}

// ---------------- driver ----------------

extern "C" void kernel_launch(void* const* d_in, const int* in_sizes, int n_in,
                              void* d_out, int out_size, void* d_ws, size_t ws_size,
                              hipStream_t stream) {
  (void)n_in; (void)out_size; (void)ws_size;
  const float* x  = (const float*)d_in[0];
  const int* ei[3] = { (const int*)d_in[1], (const int*)d_in[2], (const int*)d_in[3] };
  const float* W1[3] = { (const float*)d_in[4],  (const float*)d_in[6],  (const float*)d_in[8]  };
  const float* b1[3] = { (const float*)d_in[5],  (const float*)d_in[7],  (const float*)d_in[9]  };
  const float* W2[3] = { (const float*)d_in[10], (const float*)d_in[12], (const float*)d_in[14] };
  const float* b2[3] = { (const float*)d_in[11], (const float*)d_in[13], (const float*)d_in[15] };

  const int N = in_sizes[0] / D;
  const int E = in_sizes[1] / 2;
  float* out = (float*)d_out;

  // workspace layout (all fp32): dinv[3N] | norm[3E] | h[ND] | acc[ND] | h1[ND]
  float* ws = (float*)d_ws;
  float* dinv[3] = { ws, ws + N, ws + 2 * (size_t)N };
  float* norm[3] = { ws + 3 * (size_t)N,
                     ws + 3 * (size_t)N + E,
                     ws + 3 * (size_t)N + 2 * (size_t)E };
  float* h   = ws + 3 * (size_t)N + 3 * (size_t)E;       // N*D
  float* acc = h + (size_t)N * D;                        // N*D
  float* h1  = acc + (size_t)N * D;                      // N*D

  const int ND  = N * D;
  const int ND4 = ND / 4;
  const int gN   = (N + TPB - 1) / TPB;
  const int gE   = (E + TPB - 1) / TPB;
  const int gND4 = (ND4 + TPB - 1) / TPB;
  const int gEdge = (int)(((long long)E * 32 + TPB - 1) / TPB);
  const int ntiles = (N + 15) / 16;
  const int gGemm = (ntiles + WAVES_PER_BLOCK - 1) / WAVES_PER_BLOCK;

  // symmetric normalization terms (shared by both layers)
  k_init_deg<<<gN, TPB, 0, stream>>>(dinv[0], dinv[1], dinv[2], N);
  for (int r = 0; r < 3; ++r)
    k_count_deg<<<gE, TPB, 0, stream>>>(ei[r], E, dinv[r]);
  k_dinv<<<gN, TPB, 0, stream>>>(dinv[0], dinv[1], dinv[2], N);
  for (int r = 0; r < 3; ++r)
    k_norm<<<gE, TPB, 0, stream>>>(ei[r], E, dinv[r], norm[r]);

  // ---- layer 1 ----
  k_zero4<<<gND4, TPB, 0, stream>>>((float4*)acc, ND4);
  for (int r = 0; r < 3; ++r) {
    k_gemm_wmma<<<gGemm, TPB, 0, stream>>>(x, W1[r], dinv[r], h, acc, N);
    k_scatter<<<gEdge, TPB, 0, stream>>>(ei[r], E, h, norm[r], acc);
  }
  k_finalize1<<<gND4, TPB, 0, stream>>>(x, acc, b1[0], b1[1], b1[2], h1, ND4);

  // ---- layer 2 ----
  k_zero4<<<gND4, TPB, 0, stream>>>((float4*)acc, ND4);
  for (int r = 0; r < 3; ++r) {
    k_gemm_wmma<<<gGemm, TPB, 0, stream>>>(h1, W2[r], dinv[r], h, acc, N);
    k_scatter<<<gEdge, TPB, 0, stream>>>(ei[r], E, h, norm[r], acc);
  }
  k_finalize2<<<gND4, TPB, 0, stream>>>(acc, b2[0], b2[1], b2[2], out, ND4);
}